// Model_10282151706797
// MI455X (gfx1250) — compile-verified
//
#include <hip/hip_runtime.h>

typedef float v2f __attribute__((ext_vector_type(2)));
typedef float v8f __attribute__((ext_vector_type(8)));

// ---------------- utility kernels ----------------

__global__ void gcn_fill1(float* __restrict__ p, int n) {
  int i = blockIdx.x * blockDim.x + threadIdx.x;
  if (i < n) p[i] = 1.0f;
}

__global__ void gcn_zero(float* __restrict__ p, size_t n) {
  size_t i = (size_t)blockIdx.x * blockDim.x + threadIdx.x;
  if (i < n) p[i] = 0.0f;
}

__global__ void gcn_deg_accum(const int* __restrict__ dst, int E,
                              float* __restrict__ deg) {
  int e = blockIdx.x * blockDim.x + threadIdx.x;
  if (e < E) atomicAdd(&deg[dst[e]], 1.0f);
}

__global__ void gcn_rsqrt(const float* __restrict__ deg,
                          float* __restrict__ dis, int n) {
  int i = blockIdx.x * blockDim.x + threadIdx.x;
  if (i < n) dis[i] = rsqrtf(deg[i]);   // deg >= 1 always (self loop)
}

__global__ void gcn_bias_act(float* __restrict__ x, const float* __restrict__ b,
                             size_t n, int cmask, int relu) {
  size_t i = (size_t)blockIdx.x * blockDim.x + threadIdx.x;
  if (i >= n) return;
  float v = x[i] + b[i & (size_t)cmask];
  x[i] = relu ? fmaxf(v, 0.0f) : v;
}

// ---------------- f32 WMMA GEMM: C[M,N] = A[M,K] * B[K,N] ----------------
// One wave (32 lanes) produces one 16x16 tile of C via V_WMMA_F32_16X16X4_F32.
// A fragment (16x4): lane L holds row (L&15); VGPR0/1 hold K = kh, kh+1 where
// kh = (L>=16) ? 2 : 0. B fragment (4x16): lane L holds col (L&15), same K split.
// C/D: VGPR v -> row (v + 8*(L>=16)), col (L&15).
__global__ void __launch_bounds__(32)
gcn_gemm_wmma_f32(const float* __restrict__ A, const float* __restrict__ B,
                  float* __restrict__ C, int M, int K, int N) {
  const int lane = threadIdx.x;
  const int mt = blockIdx.x << 4;
  const int nt = blockIdx.y << 4;
  const int r  = lane & 15;
  const int kh = (lane >> 4) << 1;             // 0 or 2

  const int row = (mt + r < M) ? (mt + r) : (M - 1);   // clamp (EXEC all-1s)
  const float* __restrict__ Arow = A + (size_t)row * K;
  const float* __restrict__ Bcol = B + nt + r;

  v8f c = {};
  for (int k = 0; k < K; k += 4) {
    v2f a, b;
    a[0] = Arow[k + kh];
    a[1] = Arow[k + kh + 1];
    b[0] = Bcol[(size_t)(k + kh) * N];
    b[1] = Bcol[(size_t)(k + kh + 1) * N];
    c = __builtin_amdgcn_wmma_f32_16x16x4_f32(false, a, false, b,
                                              (short)0, c, false, false);
  }

  const int rbase = mt + ((lane >> 4) << 3);
  const int col   = nt + r;
#pragma unroll
  for (int v = 0; v < 8; ++v) {
    if (rbase + v < M) C[(size_t)(rbase + v) * N + col] = c[v];
  }
}

// ---------------- edge gather/scale/scatter-add ----------------
// e in [0,E): real edges; e in [E, E+Nn): self loops (src=dst=e-E).
template <int CH, int EPB>
__global__ void __launch_bounds__(CH * EPB)
gcn_scatter(const float* __restrict__ h, const int* __restrict__ src,
            const int* __restrict__ dst, const float* __restrict__ dis,
            float* __restrict__ out, int E, int Nn) {
  int e = blockIdx.x * EPB + threadIdx.y;
  if (e >= E + Nn) return;
  int s, d;
  if (e < E) { s = src[e]; d = dst[e]; }
  else       { s = e - E;  d = s; }
  const float nrm = dis[s] * dis[d];
  const int c = threadIdx.x;
  atomicAdd(&out[(size_t)d * CH + c], h[(size_t)s * CH + c] * nrm);
}

// ---------------- driver ----------------

extern "C" void kernel_launch(void* const* d_in, const int* in_sizes, int n_in,
                              void* d_out, int out_size, void* d_ws, size_t ws_size,
                              hipStream_t stream) {
  (void)n_in; (void)out_size; (void)ws_size;

  const float* x  = (const float*)d_in[0];
  const int*   ei = (const int*)d_in[1];   // [2, E] int32
  const float* W1 = (const float*)d_in[2];
  const float* b1 = (const float*)d_in[3];
  const float* W2 = (const float*)d_in[4];
  const float* b2 = (const float*)d_in[5];
  float* out = (float*)d_out;

  const int IN = 128, HID = 256, OUT = 64;
  const int N = in_sizes[0] / IN;          // 100000
  const int E = in_sizes[1] / 2;           // 500000
  const int* src = ei;
  const int* dst = ei + E;

  float* deg  = (float*)d_ws;
  float* dis  = deg + N;
  float* h1   = dis + N;
  float* agg1 = h1 + (size_t)N * HID;
  float* h2   = agg1 + (size_t)N * HID;

  const size_t n1 = (size_t)N * HID;
  const size_t n2 = (size_t)N * OUT;

  // symmetric normalization: deg (incl. self loop) -> dis = rsqrt(deg)
  gcn_fill1<<<(N + 255) / 256, 256, 0, stream>>>(deg, N);
  gcn_deg_accum<<<(E + 255) / 256, 256, 0, stream>>>(dst, E, deg);
  gcn_rsqrt<<<(N + 255) / 256, 256, 0, stream>>>(deg, dis, N);

  // zero the scatter accumulators (ws/d_out are poisoned by harness)
  gcn_zero<<<(int)((n1 + 255) / 256), 256, 0, stream>>>(agg1, n1);
  gcn_zero<<<(int)((n2 + 255) / 256), 256, 0, stream>>>(out, n2);

  // ---- layer 1: h1 = x @ W1 ; agg1 = scatter(norm * h1[src]) ; relu(+b1) ----
  gcn_gemm_wmma_f32<<<dim3((N + 15) / 16, HID / 16), 32, 0, stream>>>(
      x, W1, h1, N, IN, HID);
  gcn_scatter<256, 1><<<E + N, dim3(256, 1), 0, stream>>>(
      h1, src, dst, dis, agg1, E, N);
  gcn_bias_act<<<(int)((n1 + 255) / 256), 256, 0, stream>>>(
      agg1, b1, n1, HID - 1, 1);

  // ---- layer 2: h2 = agg1 @ W2 ; out = scatter(norm * h2[src]) + b2 ----
  gcn_gemm_wmma_f32<<<dim3((N + 15) / 16, OUT / 16), 32, 0, stream>>>(
      agg1, W2, h2, N, HID, OUT);
  gcn_scatter<64, 4><<<(E + N + 3) / 4, dim3(64, 4), 0, stream>>>(
      h2, src, dst, dis, out, E, N);
  gcn_bias_act<<<(int)((n2 + 255) / 256), 256, 0, stream>>>(
      out, b2, n2, OUT - 1, 0);
}